// VNAgg_14242111554125
// MI455X (gfx1250) — compile-verified
//
#include <hip/hip_runtime.h>

typedef __attribute__((ext_vector_type(2))) float v2f;
typedef __attribute__((ext_vector_type(4))) float v4f;
typedef __attribute__((ext_vector_type(8))) float v8f;

#define DIM 128
#define D2  256
#define EPS 1e-5f

// ---------------------------------------------------------------------------
// K1: deterministic segment-sum + virtual node add.
// One block (128 thr = 4 waves) per graph; batch_vector is sorted so each
// graph's nodes are contiguous. Lane l of every wave owns columns 4l..4l+3
// (float4 -> global_load_b128: one 512B row per wave-instruction). The node
// range is split across the 4 waves (fixed partition), partials combined in
// fixed order via LDS -> bitwise deterministic.
// ---------------------------------------------------------------------------
__global__ void vn_segsum(const float* __restrict__ emb,
                          const float* __restrict__ vnode,
                          const int*   __restrict__ batch,
                          float* __restrict__ h,
                          int n_nodes) {
    const int g    = blockIdx.x;
    const int tid  = threadIdx.x;       // 0..127
    const int wave = tid >> 5;          // 0..3
    const int lane = tid & 31;          // columns 4*lane .. 4*lane+3

    // lower_bound(batch, g) and lower_bound(batch, g+1)
    int lo = 0, hi = n_nodes;
    while (lo < hi) { int mid = (lo + hi) >> 1; if (batch[mid] < g) lo = mid + 1; else hi = mid; }
    const int start = lo;
    hi = n_nodes;
    while (lo < hi) { int mid = (lo + hi) >> 1; if (batch[mid] < g + 1) lo = mid + 1; else hi = mid; }
    const int end = lo;

    // Fixed 4-way split of this graph's node range across waves.
    const int cnt   = end - start;
    const int chunk = (cnt + 3) >> 2;
    const int s0    = start + wave * chunk;
    const int s1    = min(s0 + chunk, end);

    // 2 independent accumulators per thread for memory-level parallelism.
    v4f a0 = {0.f, 0.f, 0.f, 0.f};
    v4f a1 = {0.f, 0.f, 0.f, 0.f};
    const v4f* p = (const v4f*)emb + (size_t)s0 * 32 + lane;
    int n = s0;
    for (; n + 2 <= s1; n += 2, p += 64) {
        v4f x0 = p[0];
        v4f x1 = p[32];
        a0.x += x0.x; a0.y += x0.y; a0.z += x0.z; a0.w += x0.w;
        a1.x += x1.x; a1.y += x1.y; a1.z += x1.z; a1.w += x1.w;
    }
    if (n < s1) {
        v4f x0 = p[0];
        a0.x += x0.x; a0.y += x0.y; a0.z += x0.z; a0.w += x0.w;
    }
    a0.x += a1.x; a0.y += a1.y; a0.z += a1.z; a0.w += a1.w;

    // Fixed-order combine across waves via LDS.
    __shared__ v4f part[4][32];
    part[wave][lane] = a0;
    __syncthreads();
    if (wave == 0) {
        v4f r = ((const v4f*)vnode)[(size_t)g * 32 + lane];
        #pragma unroll
        for (int w = 0; w < 4; ++w) {
            v4f x = part[w][lane];
            r.x += x.x; r.y += x.y; r.z += x.z; r.w += x.w;
        }
        ((v4f*)h)[(size_t)g * 32 + lane] = r;
    }
}

// ---------------------------------------------------------------------------
// K2/K5: f32 WMMA GEMM, C[M,N] = A[M,K] * W[N,K]^T  (wave-per-16x16-tile).
// FUSE=true applies relu(x*scale[k]+shift[k]) to A elements on load
// (folds the previous BatchNorm+ReLU into the GEMM).
// Uses V_WMMA_F32_16X16X4_F32, K stepped by 4.
// ---------------------------------------------------------------------------
template <bool FUSE>
__global__ void gemm_wmma_f32(const float* __restrict__ A,
                              const float* __restrict__ W,
                              float* __restrict__ C,
                              const float* __restrict__ scale,
                              const float* __restrict__ shift,
                              int M, int N, int K) {
    const int wave   = (blockIdx.x * blockDim.x + threadIdx.x) >> 5;
    const int lane   = threadIdx.x & 31;
    const int tilesN = N >> 4;
    const int m0     = (wave / tilesN) << 4;
    const int n0     = (wave % tilesN) << 4;
    if (m0 >= M) return;                 // wave-uniform: EXEC stays all-ones

    const int half = lane >> 4;          // 0: K={0,1}, 1: K={2,3}
    const int l    = lane & 15;

    v8f acc = {};
    const float* arow = A + (size_t)(m0 + l) * K + 2 * half;
    const float* brow = W + (size_t)(n0 + l) * K + 2 * half;   // B[k][n] = W[n*K + k]

    for (int k0 = 0; k0 < K; k0 += 4) {
        v2f a = *(const v2f*)(arow + k0);
        v2f b = *(const v2f*)(brow + k0);
        if (FUSE) {
            const int kk = k0 + 2 * half;
            v2f sc = *(const v2f*)(scale + kk);
            v2f sh = *(const v2f*)(shift + kk);
            a.x = fmaxf(0.f, a.x * sc.x + sh.x);
            a.y = fmaxf(0.f, a.y * sc.y + sh.y);
        }
        // 8 args: (neg_a, A, neg_b, B, c_mod, C, reuse_a, reuse_b)
        acc = __builtin_amdgcn_wmma_f32_16x16x4_f32(
                  false, a, false, b, (short)0, acc, false, false);
    }

    // C/D layout: lanes 0-15 -> M = m0..m0+7, lanes 16-31 -> M = m0+8..m0+15
    float* crow = C + (size_t)(m0 + half * 8) * N + n0 + l;
#pragma unroll
    for (int i = 0; i < 8; ++i) crow[(size_t)i * N] = acc[i];
}

// ---------------------------------------------------------------------------
// K3/K6 stage 1: per-column partial sum / sum-of-squares over a row chunk.
// blockDim.x == C (128 or 256), coalesced loads, deterministic.
// ---------------------------------------------------------------------------
__global__ void col_partial(const float* __restrict__ X,
                            float* __restrict__ ps, float* __restrict__ pq,
                            int rows_per_block, int C) {
    const int c  = threadIdx.x;
    const int r0 = blockIdx.x * rows_per_block;
    const float* p = X + (size_t)r0 * C + c;
    float s = 0.f, q = 0.f;
    for (int i = 0; i < rows_per_block; ++i) {
        float x = p[(size_t)i * C];
        s += x; q += x * x;
    }
    ps[blockIdx.x * C + c] = s;
    pq[blockIdx.x * C + c] = q;
}

// Stage 2: combine partials, emit fused BN scale/shift (bias cancels in BN).
__global__ void col_finalize(const float* __restrict__ ps, const float* __restrict__ pq,
                             const float* __restrict__ gamma, const float* __restrict__ beta,
                             float* __restrict__ scale, float* __restrict__ shift,
                             int NB, int C, float invR) {
    const int c = threadIdx.x;
    float s = 0.f, q = 0.f;
    for (int b = 0; b < NB; ++b) { s += ps[b * C + c]; q += pq[b * C + c]; }
    const float m  = s * invR;
    const float v  = q * invR - m * m;    // biased variance, matches reference
    const float sc = gamma[c] * rsqrtf(v + EPS);
    scale[c] = sc;
    shift[c] = beta[c] - m * sc;
}

// K7: final BN + ReLU elementwise into d_out.
__global__ void bnrelu_out(const float* __restrict__ X,
                           const float* __restrict__ scale,
                           const float* __restrict__ shift,
                           float* __restrict__ out, int total) {
    const int i = blockIdx.x * blockDim.x + threadIdx.x;
    if (i < total) {
        const int c = i & (DIM - 1);
        out[i] = fmaxf(0.f, X[i] * scale[c] + shift[c]);
    }
}

// ---------------------------------------------------------------------------
extern "C" void kernel_launch(void* const* d_in, const int* in_sizes, int n_in,
                              void* d_out, int out_size, void* d_ws, size_t ws_size,
                              hipStream_t stream) {
    const float* vnode = (const float*)d_in[0];
    const float* emb   = (const float*)d_in[1];
    const float* W1    = (const float*)d_in[2];
    // d_in[3] = b1 (cancels under BatchNorm -> unused, exact)
    const float* g1    = (const float*)d_in[4];
    const float* beta1 = (const float*)d_in[5];
    const float* W2    = (const float*)d_in[6];
    // d_in[7] = b2 (cancels under BatchNorm -> unused, exact)
    const float* g2    = (const float*)d_in[8];
    const float* beta2 = (const float*)d_in[9];
    const int*   batch = (const int*)d_in[10];

    const int B  = in_sizes[0] / DIM;   // 8192 graphs
    const int nN = in_sizes[1] / DIM;   // 1,000,000 nodes

    float* ws     = (float*)d_ws;
    float* h      = ws;                          // [B, 128]
    float* h1     = h  + (size_t)B * DIM;        // [B, 256]
    float* h2     = h1 + (size_t)B * D2;         // [B, 128]
    const int NB  = 64;                          // partial-reduction blocks
    float* ps     = h2 + (size_t)B * DIM;        // [NB, 256] partial sums
    float* pq     = ps + (size_t)NB * D2;        // [NB, 256] partial sumsq
    float* scale1 = pq + (size_t)NB * D2;        // [256]
    float* shift1 = scale1 + D2;                 // [256]
    float* scale2 = shift1 + D2;                 // [128]
    float* shift2 = scale2 + DIM;                // [128]

    // 1) h = virtual_node + segment_sum(embeddings)   (BW-dominant, ~512 MB)
    vn_segsum<<<B, DIM, 0, stream>>>(emb, vnode, batch, h, nN);

    // 2) h1 = h @ W1^T            (M=8192, N=256, K=128)
    {
        const int tiles = (B / 16) * (D2 / 16);
        gemm_wmma_f32<false><<<tiles / 8, 256, 0, stream>>>(
            h, W1, h1, nullptr, nullptr, B, D2, DIM);
    }

    // 3) BN1 stats -> scale1/shift1
    col_partial <<<NB, D2, 0, stream>>>(h1, ps, pq, B / NB, D2);
    col_finalize<<<1,  D2, 0, stream>>>(ps, pq, g1, beta1, scale1, shift1,
                                        NB, D2, 1.0f / (float)B);

    // 4) h2 = relu(bn1(h1)) @ W2^T   (BN+ReLU fused into A-load; M=8192,N=128,K=256)
    {
        const int tiles = (B / 16) * (DIM / 16);
        gemm_wmma_f32<true><<<tiles / 8, 256, 0, stream>>>(
            h1, W2, h2, scale1, shift1, B, DIM, D2);
    }

    // 5) BN2 stats -> scale2/shift2
    col_partial <<<NB, DIM, 0, stream>>>(h2, ps, pq, B / NB, DIM);
    col_finalize<<<1,  DIM, 0, stream>>>(ps, pq, g2, beta2, scale2, shift2,
                                         NB, DIM, 1.0f / (float)B);

    // 6) out = relu(bn2(h2))
    const int total = B * DIM;
    bnrelu_out<<<(total + 255) / 256, 256, 0, stream>>>(
        h2, scale2, shift2, (float*)d_out, total);
}